// NTM_37572373905640
// MI455X (gfx1250) — compile-verified
//
#include <hip/hip_runtime.h>
#include <math.h>

#define B_   256
#define N_   1024
#define M_   256
#define H_   256
#define G3_  768      // 3*H
#define CO_  1292     // 5*M+12
#define EPSF 1e-12f

typedef float v2f __attribute__((ext_vector_type(2)));
typedef float v4f __attribute__((ext_vector_type(4)));
typedef float v8f __attribute__((ext_vector_type(8)));

__device__ __forceinline__ float sigmoidf_(float x) { return 1.f / (1.f + expf(-x)); }
__device__ __forceinline__ float softplusf_(float x) { return x > 20.f ? x : log1pf(expf(x)); }

// ---------------------------------------------------------------------------
// Pass 1: read_vec partial sums + per-row L2 norms of memory. One full read.
// grid (B, 4) x 256 threads (8 waves). Wave handles rows n = chunk*256 + wave + 8i.
// Lane owns 8 contiguous m (two float4 loads per row).
// ---------------------------------------------------------------------------
__global__ void k_pass1(const float* __restrict__ mem, const float* __restrict__ rw,
                        float* __restrict__ norms, float* __restrict__ rvpart) {
    const int b = blockIdx.x;
    const int chunk = blockIdx.y;
    const int lane = threadIdx.x & 31;
    const int wave = threadIdx.x >> 5;
    const int m0 = lane * 8;
    const float* memb = mem + (size_t)b * N_ * M_;
    const int nbeg = chunk * (N_ / 4);
    const int nend = nbeg + (N_ / 4);

    float rv[8];
#pragma unroll
    for (int j = 0; j < 8; ++j) rv[j] = 0.f;

    for (int n = nbeg + wave; n < nend; n += 8) {
        const float* row = memb + (size_t)n * M_ + m0;
        if (n + 8 < nend) __builtin_prefetch(memb + (size_t)(n + 8) * M_ + m0, 0, 0);
        v4f v0 = *(const v4f*)(row);
        v4f v1 = *(const v4f*)(row + 4);
        float w = rw[(size_t)b * N_ + n];
        rv[0] += w * v0[0]; rv[1] += w * v0[1]; rv[2] += w * v0[2]; rv[3] += w * v0[3];
        rv[4] += w * v1[0]; rv[5] += w * v1[1]; rv[6] += w * v1[2]; rv[7] += w * v1[3];
        float sq = v0[0]*v0[0] + v0[1]*v0[1] + v0[2]*v0[2] + v0[3]*v0[3]
                 + v1[0]*v1[0] + v1[1]*v1[1] + v1[2]*v1[2] + v1[3]*v1[3];
#pragma unroll
        for (int off = 16; off; off >>= 1) sq += __shfl_xor(sq, off, 32);
        if (lane == 0) norms[(size_t)b * N_ + n] = sqrtf(sq);
    }

    __shared__ float lds[8 * 256];
#pragma unroll
    for (int j = 0; j < 8; ++j) lds[wave * 256 + m0 + j] = rv[j];
    __syncthreads();
    const int t = threadIdx.x;
    float s = 0.f;
#pragma unroll
    for (int w = 0; w < 8; ++w) s += lds[w * 256 + t];
    rvpart[((size_t)b * 4 + chunk) * M_ + t] = s;
}

__global__ void k_rv_combine(const float* __restrict__ part, float* __restrict__ rv) {
    int i = blockIdx.x * blockDim.x + threadIdx.x;   // B*M
    int b = i >> 8, m = i & 255;
    rv[i] = part[((size_t)b*4 + 0)*M_ + m] + part[((size_t)b*4 + 1)*M_ + m]
          + part[((size_t)b*4 + 2)*M_ + m] + part[((size_t)b*4 + 3)*M_ + m];
}

// ---------------------------------------------------------------------------
// fp32 WMMA GEMM: C[B_ x N] = A[B_ x K] * W[N x K]^T + bias[N]
// One wave per 16x16 tile, V_WMMA_F32_16X16X4_F32, K stepped by 4.
// ---------------------------------------------------------------------------
__global__ void k_gemm_wmma(const float* __restrict__ A, const float* __restrict__ W,
                            const float* __restrict__ bias, float* __restrict__ C,
                            int Ncols, int K) {
    const int m0 = blockIdx.x * 16;
    const int n0 = blockIdx.y * 16;
    const int lane = threadIdx.x;
    const int half = lane >> 4;
    const int r = lane & 15;
    const float* arow = A + (size_t)(m0 + r) * K;
    int wn = n0 + r; if (wn >= Ncols) wn = Ncols - 1;   // clamp (store masked below)
    const float* wrow = W + (size_t)wn * K;

    v8f acc = {};
    for (int k0 = 0; k0 < K; k0 += 4) {
        v2f a, bb;
        a[0]  = arow[k0 + 2 * half];
        a[1]  = arow[k0 + 1 + 2 * half];
        bb[0] = wrow[k0 + 2 * half];
        bb[1] = wrow[k0 + 1 + 2 * half];
        acc = __builtin_amdgcn_wmma_f32_16x16x4_f32(false, a, false, bb,
                                                    (short)0, acc, false, false);
    }
    if (n0 + r < Ncols) {
        float bv = bias[n0 + r];
#pragma unroll
        for (int i = 0; i < 8; ++i) {
            int mm = m0 + i + 8 * half;
            C[(size_t)mm * Ncols + n0 + r] = acc[i] + bv;
        }
    }
}

// ---------------------------------------------------------------------------
// GRU elementwise combine
// ---------------------------------------------------------------------------
__global__ void k_gru(const float* __restrict__ gi, const float* __restrict__ gh,
                      const float* __restrict__ hprev, float* __restrict__ hnew) {
    int i = blockIdx.x * blockDim.x + threadIdx.x;   // B*H
    int b = i >> 8, h = i & 255;
    const float* gib = gi + (size_t)b * G3_;
    const float* ghb = gh + (size_t)b * G3_;
    float r = sigmoidf_(gib[h]       + ghb[h]);
    float z = sigmoidf_(gib[256 + h] + ghb[256 + h]);
    float n = tanhf(gib[512 + h] + r * ghb[512 + h]);
    hnew[i] = (1.f - z) * n + z * hprev[i];
}

// ---------------------------------------------------------------------------
// Head prep: per-b key norms, scaled keys (beta/(|k|+eps) folded in),
// g / shift-softmax / gamma params, erase/add vectors, y output.
// ---------------------------------------------------------------------------
__global__ void k_prep(const float* __restrict__ co, float* __restrict__ skr,
                       float* __restrict__ skw, float* __restrict__ params,
                       float* __restrict__ erasev, float* __restrict__ addv,
                       float* __restrict__ y_out) {
    const int b = blockIdx.x, t = threadIdx.x;
    const float* c = co + (size_t)b * CO_;
    __shared__ float red[256];

    // read head key norm
    float kr = c[t];
    red[t] = kr * kr; __syncthreads();
    for (int s = 128; s; s >>= 1) { if (t < s) red[t] += red[t + s]; __syncthreads(); }
    float nr = sqrtf(red[0]); __syncthreads();
    float betar = softplusf_(c[256]);
    skr[(size_t)b * M_ + t] = kr * betar / (nr + EPSF);

    // write head key norm
    float kw = c[262 + t];
    red[t] = kw * kw; __syncthreads();
    for (int s = 128; s; s >>= 1) { if (t < s) red[t] += red[t + s]; __syncthreads(); }
    float nw = sqrtf(red[0]);
    float betaw = softplusf_(c[518]);
    skw[(size_t)b * M_ + t] = kw * betaw / (nw + EPSF);

    if (t == 0) {
        // read head: gate c[257], shift c[258..260], gamma c[261]
        float g = sigmoidf_(c[257]);
        float a0 = c[258], a1 = c[259], a2 = c[260];
        float mx = fmaxf(a0, fmaxf(a1, a2));
        float e0 = expf(a0 - mx), e1 = expf(a1 - mx), e2 = expf(a2 - mx);
        float inv = 1.f / (e0 + e1 + e2);
        float* p = params + b * 16;
        p[0] = g; p[1] = e0 * inv; p[2] = e1 * inv; p[3] = e2 * inv;
        p[4] = 1.f + softplusf_(c[261]);
        // write head: gate c[519], shift c[520..522], gamma c[523]
        g = sigmoidf_(c[519]);
        a0 = c[520]; a1 = c[521]; a2 = c[522];
        mx = fmaxf(a0, fmaxf(a1, a2));
        e0 = expf(a0 - mx); e1 = expf(a1 - mx); e2 = expf(a2 - mx);
        inv = 1.f / (e0 + e1 + e2);
        p[8] = g; p[9] = e0 * inv; p[10] = e1 * inv; p[11] = e2 * inv;
        p[12] = 1.f + softplusf_(c[523]);
    }
    erasev[(size_t)b * M_ + t] = sigmoidf_(c[524 + t]);
    addv[(size_t)b * M_ + t]   = c[780 + t];
    y_out[(size_t)b * M_ + t]  = c[1036 + t];
}

// ---------------------------------------------------------------------------
// Pass 2: cosine scores for BOTH heads in one memory read.
// sr/sw[b,n] = beta * cos (beta & 1/|key| folded into scaled keys).
// ---------------------------------------------------------------------------
__global__ void k_cos(const float* __restrict__ mem, const float* __restrict__ norms,
                      const float* __restrict__ skr, const float* __restrict__ skw,
                      float* __restrict__ sr, float* __restrict__ sw) {
    const int b = blockIdx.x;
    const int chunk = blockIdx.y;
    const int lane = threadIdx.x & 31;
    const int wave = threadIdx.x >> 5;
    const int m0 = lane * 8;
    const float* memb = mem + (size_t)b * N_ * M_;
    const int nbeg = chunk * (N_ / 4);
    const int nend = nbeg + (N_ / 4);

    float kr[8], kw[8];
#pragma unroll
    for (int j = 0; j < 8; ++j) {
        kr[j] = skr[(size_t)b * M_ + m0 + j];
        kw[j] = skw[(size_t)b * M_ + m0 + j];
    }
    for (int n = nbeg + wave; n < nend; n += 8) {
        const float* row = memb + (size_t)n * M_ + m0;
        if (n + 8 < nend) __builtin_prefetch(memb + (size_t)(n + 8) * M_ + m0, 0, 0);
        v4f v0 = *(const v4f*)(row);
        v4f v1 = *(const v4f*)(row + 4);
        float dr = v0[0]*kr[0] + v0[1]*kr[1] + v0[2]*kr[2] + v0[3]*kr[3]
                 + v1[0]*kr[4] + v1[1]*kr[5] + v1[2]*kr[6] + v1[3]*kr[7];
        float dw = v0[0]*kw[0] + v0[1]*kw[1] + v0[2]*kw[2] + v0[3]*kw[3]
                 + v1[0]*kw[4] + v1[1]*kw[5] + v1[2]*kw[6] + v1[3]*kw[7];
#pragma unroll
        for (int off = 16; off; off >>= 1) {
            dr += __shfl_xor(dr, off, 32);
            dw += __shfl_xor(dw, off, 32);
        }
        if (lane == 0) {
            float inv = 1.f / (norms[(size_t)b * N_ + n] + EPSF);
            sr[(size_t)b * N_ + n] = dr * inv;
            sw[(size_t)b * N_ + n] = dw * inv;
        }
    }
}

// ---------------------------------------------------------------------------
// Addressing finish: softmax -> interpolate -> circular shift -> sharpen.
// grid (B, 2 heads) x 256 threads; N=1024 held in LDS (4 elems/thread).
// ---------------------------------------------------------------------------
__global__ void k_addr(const float* __restrict__ scores_r, const float* __restrict__ scores_w,
                       const float* __restrict__ prev_r, const float* __restrict__ prev_w,
                       const float* __restrict__ params, float* __restrict__ out_r,
                       float* __restrict__ out_w, float* __restrict__ wtw_ws) {
    const int b = blockIdx.x, head = blockIdx.y, t = threadIdx.x;
    const float* sc = (head == 0 ? scores_r : scores_w) + (size_t)b * N_;
    const float* pv = (head == 0 ? prev_r : prev_w) + (size_t)b * N_;
    const float* p = params + b * 16 + head * 8;
    const float g = p[0], s0 = p[1], s1 = p[2], s2 = p[3], gamma = p[4];

    __shared__ float wt[N_];
    __shared__ float red[256];

    float v[4];
    float mx = -3.4e38f;
#pragma unroll
    for (int j = 0; j < 4; ++j) { v[j] = sc[t + 256 * j]; mx = fmaxf(mx, v[j]); }
    red[t] = mx; __syncthreads();
    for (int s = 128; s; s >>= 1) { if (t < s) red[t] = fmaxf(red[t], red[t + s]); __syncthreads(); }
    mx = red[0]; __syncthreads();

    float lsum = 0.f;
#pragma unroll
    for (int j = 0; j < 4; ++j) { float e = expf(v[j] - mx); wt[t + 256 * j] = e; lsum += e; }
    red[t] = lsum; __syncthreads();
    for (int s = 128; s; s >>= 1) { if (t < s) red[t] += red[t + s]; __syncthreads(); }
    float inv = 1.f / red[0]; __syncthreads();

#pragma unroll
    for (int j = 0; j < 4; ++j) {
        int n = t + 256 * j;
        wt[n] = g * wt[n] * inv + (1.f - g) * pv[n];
    }
    __syncthreads();

    float pw[4];
    float l2 = 0.f;
#pragma unroll
    for (int j = 0; j < 4; ++j) {
        int n = t + 256 * j;
        float sh = s0 * wt[(n + 1) & (N_ - 1)] + s1 * wt[n] + s2 * wt[(n - 1) & (N_ - 1)];
        float pp = powf(sh + EPSF, gamma);
        pw[j] = pp; l2 += pp;
    }
    red[t] = l2; __syncthreads();
    for (int s = 128; s; s >>= 1) { if (t < s) red[t] += red[t + s]; __syncthreads(); }
    float invt = 1.f / (red[0] + EPSF);

    float* dst = (head == 0 ? out_r : out_w) + (size_t)b * N_;
#pragma unroll
    for (int j = 0; j < 4; ++j) {
        int n = t + 256 * j;
        float val = pw[j] * invt;
        dst[n] = val;
        if (head == 1) wtw_ws[(size_t)b * N_ + n] = val;
    }
}

// ---------------------------------------------------------------------------
// Pass 3: memory update. 1 read + 1 write, non-temporal (no reuse after).
// grid (B, N/4) x 256 threads; thread handles 4 floats of one row.
// ---------------------------------------------------------------------------
__global__ void k_update(const float* __restrict__ mem, const float* __restrict__ wt,
                         const float* __restrict__ er, const float* __restrict__ av,
                         float* __restrict__ outmem) {
    const int b = blockIdx.x;
    const int n = blockIdx.y * 4 + (threadIdx.x >> 6);
    const int m = (threadIdx.x & 63) * 4;
    const size_t idx = ((size_t)b * N_ + n) * M_ + m;
    v4f mv = __builtin_nontemporal_load((const v4f*)(mem + idx));
    float w = wt[(size_t)b * N_ + n];
    v4f e = *(const v4f*)(er + (size_t)b * M_ + m);
    v4f a = *(const v4f*)(av + (size_t)b * M_ + m);
    v4f o;
    o[0] = mv[0] * (1.f - w * e[0]) + w * a[0];
    o[1] = mv[1] * (1.f - w * e[1]) + w * a[1];
    o[2] = mv[2] * (1.f - w * e[2]) + w * a[2];
    o[3] = mv[3] * (1.f - w * e[3]) + w * a[3];
    __builtin_nontemporal_store(o, (v4f*)(outmem + idx));
}

// ---------------------------------------------------------------------------
extern "C" void kernel_launch(void* const* d_in, const int* in_sizes, int n_in,
                              void* d_out, int out_size, void* d_ws, size_t ws_size,
                              hipStream_t stream) {
    const float* x      = (const float*)d_in[0];
    const float* memory = (const float*)d_in[1];
    const float* read_w = (const float*)d_in[2];
    const float* write_w= (const float*)d_in[3];
    const float* hidden = (const float*)d_in[4];
    const float* w_ih   = (const float*)d_in[5];
    const float* w_hh   = (const float*)d_in[6];
    const float* b_ih   = (const float*)d_in[7];
    const float* b_hh   = (const float*)d_in[8];
    const float* proj_w = (const float*)d_in[9];
    const float* proj_b = (const float*)d_in[10];

    float* out = (float*)d_out;
    float* y_out      = out;                              // B*M
    float* newmem_out = out + (size_t)B_ * M_;            // B*N*M
    float* nrw_out    = newmem_out + (size_t)B_ * N_ * M_;// B*N
    float* nww_out    = nrw_out + (size_t)B_ * N_;        // B*N

    float* ws = (float*)d_ws;
    float* norms  = ws;                         // B*N      = 262144
    float* rvpart = norms  + (size_t)B_ * N_;   // 4*B*M    = 262144
    float* rv     = rvpart + (size_t)4 * B_ * M_;
    float* gi     = rv     + (size_t)B_ * M_;   // B*768
    float* gh     = gi     + (size_t)B_ * G3_;  // B*768
    float* h1     = gh     + (size_t)B_ * G3_;  // B*H
    float* h2     = h1     + (size_t)B_ * H_;   // B*H
    float* co     = h2     + (size_t)B_ * H_;   // B*1292
    float* skr    = co     + (size_t)B_ * CO_;  // B*M
    float* skw    = skr    + (size_t)B_ * M_;   // B*M
    float* params = skw    + (size_t)B_ * M_;   // B*16
    float* erasev = params + (size_t)B_ * 16;   // B*M
    float* addv   = erasev + (size_t)B_ * M_;   // B*M
    float* sr     = addv   + (size_t)B_ * M_;   // B*N
    float* sw     = sr     + (size_t)B_ * N_;   // B*N
    float* wtw    = sw     + (size_t)B_ * N_;   // B*N

    // Pass 1: read_vec + row norms (one full memory read)
    k_pass1<<<dim3(B_, 4), 256, 0, stream>>>(memory, read_w, norms, rvpart);
    k_rv_combine<<<(B_ * M_) / 256, 256, 0, stream>>>(rvpart, rv);

    // GRU step 1 (input = x)
    k_gemm_wmma<<<dim3(16, G3_ / 16), 32, 0, stream>>>(x, w_ih, b_ih, gi, G3_, M_);
    k_gemm_wmma<<<dim3(16, G3_ / 16), 32, 0, stream>>>(hidden, w_hh, b_hh, gh, G3_, H_);
    k_gru<<<(B_ * H_) / 256, 256, 0, stream>>>(gi, gh, hidden, h1);

    // GRU step 2 (input = read_vec)
    k_gemm_wmma<<<dim3(16, G3_ / 16), 32, 0, stream>>>(rv, w_ih, b_ih, gi, G3_, M_);
    k_gemm_wmma<<<dim3(16, G3_ / 16), 32, 0, stream>>>(h1, w_hh, b_hh, gh, G3_, H_);
    k_gru<<<(B_ * H_) / 256, 256, 0, stream>>>(gi, gh, h1, h2);

    // Projection: co = h2 @ proj_w^T + proj_b  (1292 cols -> 81 tiles)
    k_gemm_wmma<<<dim3(16, (CO_ + 15) / 16), 32, 0, stream>>>(h2, proj_w, proj_b, co, CO_, H_);

    // Head params + scaled keys + erase/add + y output
    k_prep<<<B_, 256, 0, stream>>>(co, skr, skw, params, erasev, addv, y_out);

    // Pass 2: both heads' cosine scores in one full memory read
    k_cos<<<dim3(B_, 4), 256, 0, stream>>>(memory, norms, skr, skw, sr, sw);

    // Addressing finish for both heads
    k_addr<<<dim3(B_, 2), 256, 0, stream>>>(sr, sw, read_w, write_w, params,
                                            nrw_out, nww_out, wtw);

    // Pass 3: memory update (one read + one write, non-temporal)
    k_update<<<dim3(B_, N_ / 4), 256, 0, stream>>>(memory, wtw, erasev, addv, newmem_out);
}